// PSRoIAlign_86912958202637
// MI455X (gfx1250) — compile-verified
//
#include <hip/hip_runtime.h>
#include <hip/hip_bf16.h>
#include <math.h>

#define SPATIAL_SCALE (1.0f / 16.0f)
#define ROI_SIZE      7
#define POOLED_DIM    10
#define CCH           (POOLED_DIM * ROI_SIZE * ROI_SIZE)   /* 490 channels */
#define BINS          (ROI_SIZE * ROI_SIZE)                /* 49 bins      */

// One ROI per block (512 threads = 16 wave32 waves, 490 active).
// Thread t maps directly to output element t of this ROI, and because the
// position-sensitive channel index (c*7+ph)*7+pw == t, also to channel t.
__global__ __launch_bounds__(512) void psroialign_max_kernel(
    const float* __restrict__ bottom_data,   // (B, 490, H, W)
    const float* __restrict__ bottom_rois,   // (N, 5) [bidx, x1, y1, x2, y2]
    float* __restrict__ out,                 // (N, 10, 7, 7) == (N, 490)
    int H, int W)
{
    const int n = blockIdx.x;
    const int t = threadIdx.x;

    // Uniform (blockIdx-only) addressing -> compiler lowers to scalar loads.
    const float r0  = bottom_rois[n * 5 + 0];
    const float rx1 = bottom_rois[n * 5 + 1];
    const float ry1 = bottom_rois[n * 5 + 2];
    const float rx2 = bottom_rois[n * 5 + 3];
    const float ry2 = bottom_rois[n * 5 + 4];

    if (t >= CCH) return;

    const int c   = t / BINS;
    const int bin = t - c * BINS;
    const int ph  = bin / ROI_SIZE;
    const int pw  = bin - ph * ROI_SIZE;

    const int   bidx   = (int)r0;                    // trunc toward zero, matches astype(int32)
    const float roi_sw = rx1 * SPATIAL_SCALE;
    const float roi_sh = ry1 * SPATIAL_SCALE;
    const float roi_w  = fmaxf(rx2 * SPATIAL_SCALE - roi_sw, 0.1f);
    const float roi_h  = fmaxf(ry2 * SPATIAL_SCALE - roi_sh, 0.1f);
    const float bin_h  = roi_h * (1.0f / ROI_SIZE);
    const float bin_w  = roi_w * (1.0f / ROI_SIZE);

    // Position-sensitive channel plane: cin == t.
    const float* __restrict__ plane =
        bottom_data + ((size_t)bidx * CCH + (size_t)t) * (size_t)(H * W);

    const float fH  = (float)H;
    const float fW  = (float)W;
    const float fH1 = (float)(H - 1);
    const float fW1 = (float)(W - 1);

    // ---- Pre-compute all sample coordinates (sy,sx in {0,1}) ---------------
    int   iy0[2], iy1[2], ix0[2], ix1[2];
    float wly[2], why[2], wlx[2], whx[2];
    bool  vy[2], vx[2];

#pragma unroll
    for (int s = 0; s < 2; ++s) {
        const float y = roi_sh + (float)ph * bin_h + ((float)s + 0.5f) * bin_h * 0.5f;
        vy[s] = (y > -1.0f) && (y < fH);
        const float yc = fminf(fmaxf(y, 0.0f), fH1);
        const int   y0 = (int)floorf(yc);
        iy0[s] = y0;
        iy1[s] = min(y0 + 1, H - 1);
        wly[s] = yc - (float)y0;
        why[s] = 1.0f - wly[s];

        const float x = roi_sw + (float)pw * bin_w + ((float)s + 0.5f) * bin_w * 0.5f;
        vx[s] = (x > -1.0f) && (x < fW);
        const float xc = fminf(fmaxf(x, 0.0f), fW1);
        const int   x0 = (int)floorf(xc);
        ix0[s] = x0;
        ix1[s] = min(x0 + 1, W - 1);
        wlx[s] = xc - (float)x0;
        whx[s] = 1.0f - wlx[s];
    }

    // Warm L1 for the first tap's cacheline (gfx1250 global_prefetch_b8;
    // speculative, no LOADcnt cost).
    __builtin_prefetch((const void*)(plane + iy0[0] * W + ix0[0]), 0, 3);

    // ---- Issue all 16 gathers up front (full unroll), then combine ---------
    float maxv = -INFINITY;
#pragma unroll
    for (int sy = 0; sy < 2; ++sy) {
        const int rowA = iy0[sy] * W;
        const int rowB = iy1[sy] * W;
#pragma unroll
        for (int sx = 0; sx < 2; ++sx) {
            const float v00 = plane[rowA + ix0[sx]];
            const float v01 = plane[rowA + ix1[sx]];
            const float v10 = plane[rowB + ix0[sx]];
            const float v11 = plane[rowB + ix1[sx]];
            float v = why[sy] * (whx[sx] * v00 + wlx[sx] * v01)
                    + wly[sy] * (whx[sx] * v10 + wlx[sx] * v11);
            v = (vy[sy] && vx[sx]) ? v : 0.0f;
            maxv = fmaxf(maxv, v);
        }
    }

    // Coalesced store: out layout (N, 10, 7, 7) flattened == n*490 + t.
    out[(size_t)n * CCH + t] = maxv;
}

extern "C" void kernel_launch(void* const* d_in, const int* in_sizes, int n_in,
                              void* d_out, int out_size, void* d_ws, size_t ws_size,
                              hipStream_t stream) {
    const float* bottom_data = (const float*)d_in[0];   // (B, 490, 64, 100) f32
    const float* bottom_rois = (const float*)d_in[1];   // (N, 5) f32
    float*       out         = (float*)d_out;           // (N, 10, 7, 7) f32

    const int N = in_sizes[1] / 5;          // 2048
    const int H = 64;
    const int W = 100;

    psroialign_max_kernel<<<dim3(N), dim3(512), 0, stream>>>(
        bottom_data, bottom_rois, out, H, W);
}